// MMD_VAE_69097433858313
// MI455X (gfx1250) — compile-verified
//
#include <hip/hip_runtime.h>

#define N_ROWS 8192
#define DIM 64
#define NT 512        // 8192/16 row tiles
#define NJC 4         // J chunks
#define JCHUNK (NT / NJC)
#define LSTR 68       // padded LDS row stride (floats): bank-conflict-free ds_load_b64
#define PARTIAL_OFF (2 * N_ROWS)
#define NPART (3 * NJC * NT)

typedef __attribute__((ext_vector_type(2))) float v2f;
typedef __attribute__((ext_vector_type(8))) float v8f;
typedef __attribute__((ext_vector_type(4))) int v4i;

#if defined(__gfx1250__) && __has_builtin(__builtin_amdgcn_global_load_async_to_lds_b128)
#define HAVE_ASYNC_LDS 1
typedef __attribute__((address_space(1))) v4i* gv4i_p;   // global int4*
typedef __attribute__((address_space(3))) v4i* sv4i_p;   // LDS int4*
#else
#define HAVE_ASYNC_LDS 0
#endif

// ws layout (floats): [0,8192) sq|x|^2 ; [8192,16384) sq|y|^2 ;
//                     [16384, 16384 + 3*4*512) per-(z,chunk,I) partial sums

__global__ void mmd_prep(const float* __restrict__ x, const float* __restrict__ y,
                         float* __restrict__ ws) {
    int t = blockIdx.x * blockDim.x + threadIdx.x;   // 0..16383
    const float* src = (t < N_ROWS) ? x : y;
    int row = (t < N_ROWS) ? t : (t - N_ROWS);
    const float* r = src + (size_t)row * DIM;
    float s = 0.f;
#pragma unroll
    for (int k = 0; k < DIM; k += 4) {
        float a0 = r[k], a1 = r[k + 1], a2 = r[k + 2], a3 = r[k + 3];
        s += a0 * a0 + a1 * a1 + a2 * a2 + a3 * a3;
    }
    ws[t] = s;
}

__global__ __launch_bounds__(256) void mmd_tiles(const float* __restrict__ x,
                                                 const float* __restrict__ y,
                                                 float* __restrict__ ws) {
    const int z = blockIdx.z;    // 0:(x,x) 1:(y,y) 2:(x,y)
    const int jc = blockIdx.y;   // J chunk
    const int J0 = jc * JCHUNK;
    const float* P = (z == 1) ? y : x;
    const float* Q = (z == 0) ? x : y;
    const float* sqA = ws + ((z == 1) ? N_ROWS : 0);
    const float* sqB = ws + ((z == 0) ? 0 : N_ROWS);

    const int t = threadIdx.x;
    const int lane = t & 31;
    const int wave = t >> 5;
    const int lo = lane & 15;    // M for A-frag / N for B-frag & C
    const int hi = lane >> 4;
    const int I = blockIdx.x * 8 + wave;   // each wave owns one row tile

    // staging coords: thread t copies 16B of the 16x64 B tile (row sr, cols sc..sc+3)
    const int sr = t >> 4;
    const int sc = (t & 15) * 4;

    __shared__ float bt[2][16 * LSTR];     // double-buffered B tile, padded rows

    // A tile resident in registers (WMMA f32 16x16x4 A layout)
    v2f a[16];
    const float* arow = P + (size_t)(I * 16 + lo) * DIM + 2 * hi;
#pragma unroll
    for (int kk = 0; kk < 16; ++kk)
        a[kk] = *(const v2f*)(arow + kk * 4);

    float sqa[8];
#pragma unroll
    for (int r = 0; r < 8; ++r)
        sqa[r] = sqA[I * 16 + 8 * hi + r];

    // prologue: stage tile J0 into buffer 0
    {
        const float* src = Q + (size_t)(J0 * 16 + sr) * DIM + sc;
        float4 v = *(const float4*)src;
        *(float4*)(&bt[0][sr * LSTR + sc]) = v;
    }
    __syncthreads();

    const float inv = 1.0f / 4096.0f;
    float s = 0.f;

    for (int j = 0; j < JCHUNK; ++j) {
        float* curb = &bt[j & 1][0];
        float* nxtb = &bt[(j + 1) & 1][0];
        const bool havenext = (j + 1 < JCHUNK);

        // --- prefetch tile j+1 into the alternate LDS buffer (ASYNCcnt path) ---
#if HAVE_ASYNC_LDS
        if (havenext) {
            const float* src = Q + (size_t)((J0 + j + 1) * 16 + sr) * DIM + sc;
            __builtin_amdgcn_global_load_async_to_lds_b128(
                (gv4i_p)src, (sv4i_p)(nxtb + sr * LSTR + sc), 0, 0);
        }
#else
        float4 pf;
        if (havenext)
            pf = *(const float4*)(Q + (size_t)((J0 + j + 1) * 16 + sr) * DIM + sc);
#endif

        float sqb = sqB[(J0 + j) * 16 + lo];

        // --- batched B fragments from LDS, then back-to-back WMMAs ---
        v2f b[16];
        const float* brow = curb + lo * LSTR + 2 * hi;
#pragma unroll
        for (int kk = 0; kk < 16; ++kk)
            b[kk] = *(const v2f*)(brow + kk * 4);

        v8f c = {0.f, 0.f, 0.f, 0.f, 0.f, 0.f, 0.f, 0.f};
#pragma unroll
        for (int kk = 0; kk < 16; ++kk)
            c = __builtin_amdgcn_wmma_f32_16x16x4_f32(false, a[kk], false, b[kk],
                                                      (short)0, c, false, false);

#pragma unroll
        for (int r = 0; r < 8; ++r) {
            float arg = (2.0f * c[r] - sqa[r] - sqb) * inv;   // -sqdist/dim^2
            s += __expf(arg);
        }

        // --- complete the prefetch, then barrier ---
#if HAVE_ASYNC_LDS
        if (havenext) { asm volatile("s_wait_asynccnt 0" ::: "memory"); }
#else
        if (havenext) *(float4*)(nxtb + sr * LSTR + sc) = pf;
#endif
        __syncthreads();
    }

    // intra-wave reduce (wave32); each wave owns a unique partial slot
#pragma unroll
    for (int off = 16; off > 0; off >>= 1)
        s += __shfl_xor(s, off, 32);
    if (lane == 0)
        ws[PARTIAL_OFF + (z * NJC + jc) * NT + I] = s;
}

__global__ void mmd_finalize(const float* __restrict__ ws, float* __restrict__ out) {
    const float* part = ws + PARTIAL_OFF;
    double acc[3] = {0.0, 0.0, 0.0};
    for (int z = 0; z < 3; ++z)
        for (int i = 0; i < NJC * NT; ++i)
            acc[z] += (double)part[z * NJC * NT + i];
    double m = 1.0 / ((double)N_ROWS * (double)N_ROWS);
    out[0] = (float)((acc[0] + acc[1] - 2.0 * acc[2]) * m);
}

extern "C" void kernel_launch(void* const* d_in, const int* in_sizes, int n_in,
                              void* d_out, int out_size, void* d_ws, size_t ws_size,
                              hipStream_t stream) {
    const float* x = (const float*)d_in[0];
    const float* y = (const float*)d_in[1];
    float* ws = (float*)d_ws;
    float* out = (float*)d_out;

    (void)in_sizes; (void)n_in; (void)out_size; (void)ws_size;

    mmd_prep<<<dim3((2 * N_ROWS) / 256), 256, 0, stream>>>(x, y, ws);
    mmd_tiles<<<dim3(NT / 8, NJC, 3), 256, 0, stream>>>(x, y, ws);
    mmd_finalize<<<1, 1, 0, stream>>>(ws, out);
}